// NormWeightedCompositor_73521250173219
// MI455X (gfx1250) — compile-verified
//
#include <hip/hip_runtime.h>

// Problem constants (from the reference): fragments/alphas (N=4,K=8,H=256,W=256),
// ptclds (C=64,P=100000). Output (N,C,H,W) f32.
#define K_FRAG 8
#define C_CH   64
#define HW_SZ  65536            // 256*256
#define EPS    1.0e-4f

// gfx1250 async global->LDS copy (ASYNCcnt-tracked), if the toolchain exposes it.
#if defined(__has_builtin)
#  if __has_builtin(__builtin_amdgcn_global_load_async_to_lds_b32)
#    define NWC_HAVE_ASYNC_LDS 1
#  endif
#endif
#ifndef NWC_HAVE_ASYNC_LDS
#  define NWC_HAVE_ASYNC_LDS 0
#endif

// Builtin prototype (per hipcc diagnostic): global int* (AS1), LDS int* (AS3),
// constant imm offset, constant cpol.
#define NWC_GPTR(p) ((__attribute__((address_space(1))) int*)(p))
#define NWC_LPTR(p) ((__attribute__((address_space(3))) int*)(p))

// --------------------------------------------------------------------------
// Kernel 1: transpose ptclds (C,P) -> (P,C) so each gathered feature vector
// is 256 contiguous bytes. LDS-tiled (64x64, pad 65 => conflict-free banks).
// Load phase is a global->LDS scatter: on gfx1250 we issue it as
// GLOBAL_LOAD_ASYNC_TO_LDS_B32 (per-lane LDS dest address, ASYNCcnt),
// drained with s_wait_asynccnt before the barrier.
// Bounds check hoisted out of the unrolled loop (only the last of ~1563
// blocks is partial) so the hot path is branchless.
// --------------------------------------------------------------------------
__global__ __launch_bounds__(256) void nwc_transpose_ptclds(
    const float* __restrict__ src, float* __restrict__ dst, int P)
{
    __shared__ float tile[64][65];
    const int p0  = blockIdx.x * 64;
    const int tid = threadIdx.x;

    // Load phase: lanes sweep p (coalesced 128B per instruction), 16 channels each.
    const int px = tid & 63;          // p within tile
    const int cb = (tid >> 6) * 16;   // channel base: 0,16,32,48
    const int p  = p0 + px;

    if (p < P) {
        const float* g = src + (size_t)cb * P + p;   // step by P per channel
        #pragma unroll
        for (int i = 0; i < 16; ++i) {
#if NWC_HAVE_ASYNC_LDS
            __builtin_amdgcn_global_load_async_to_lds_b32(
                NWC_GPTR(g), NWC_LPTR(&tile[px][cb + i]), /*offset=*/0, /*cpol=*/0);
#else
            tile[px][cb + i] = __builtin_nontemporal_load(g);
#endif
            g += P;
        }
    }
#if NWC_HAVE_ASYNC_LDS
    asm volatile("s_wait_asynccnt 0x0" ::: "memory");
#endif
    __syncthreads();

    // Store phase: each thread writes 16 consecutive channels of one row.
    // A wave covers 8 full rows = 2 KB contiguous, in b128 chunks.
    // Destination is the hot gather table -> temporal stores (land warm in L2).
    const int pr = tid >> 2;
    const int cq = (tid & 3) * 16;
    if (p0 + pr < P) {
        float* o = dst + (size_t)(p0 + pr) * C_CH + cq;
        #pragma unroll
        for (int j = 0; j < 4; ++j) {
            float4 v;
            v.x = tile[pr][cq + j * 4 + 0];
            v.y = tile[pr][cq + j * 4 + 1];
            v.z = tile[pr][cq + j * 4 + 2];
            v.w = tile[pr][cq + j * 4 + 3];
            *reinterpret_cast<float4*>(o + j * 4) = v;
        }
    }
}

// --------------------------------------------------------------------------
// Kernel 2: main compositor. One thread = one pixel, all 64 channels.
//  - 8 coalesced NT loads of indices + weights (read-once, protect L2 table)
//  - 8 precomputed row base pointers => gathers are b128 with immediate offsets
//  - global_prefetch_b8 of all 16 gather cachelines before the FMA loop
//  - NT b32 stores, lane-consecutive in w => one full 128B line per instruction;
//    two store bases keep every offset inside the signed 24-bit IOFFSET.
// --------------------------------------------------------------------------
__global__ __launch_bounds__(256) void nwc_composite(
    const int*   __restrict__ frags,
    const float* __restrict__ alphas,
    const float* __restrict__ tpt,     // (P, C) transposed table
    float*       __restrict__ out,
    int NHW)
{
    const int pix = blockIdx.x * blockDim.x + threadIdx.x;
    if (pix >= NHW) return;
    const int n  = pix >> 16;          // HW = 65536
    const int hw = pix & (HW_SZ - 1);

    const size_t base = ((size_t)n * K_FRAG) * HW_SZ + hw;

    const float* row[K_FRAG];
    float w[K_FRAG];
    float wsum = 0.0f;
    #pragma unroll
    for (int k = 0; k < K_FRAG; ++k) {
        const int   f = __builtin_nontemporal_load(frags  + base + (size_t)k * HW_SZ);
        const float a = __builtin_nontemporal_load(alphas + base + (size_t)k * HW_SZ);
        const bool  m = (f >= 0);
        w[k]   = m ? a : 0.0f;
        row[k] = tpt + (size_t)(m ? f : 0) * C_CH;
        wsum  += w[k];
    }

    // Kick all 8 gather targets (2 x 128B lines each) into the cache early.
    #pragma unroll
    for (int k = 0; k < K_FRAG; ++k) {
        __builtin_prefetch(row[k],      0, 3);
        __builtin_prefetch(row[k] + 32, 0, 3);
    }

    const float inv = 1.0f / fmaxf(wsum, EPS);
    float* op_lo = out + (size_t)n * C_CH * HW_SZ + hw;
    float* op_hi = op_lo + (size_t)(C_CH / 2) * HW_SZ;

    #pragma unroll
    for (int q = 0; q < C_CH / 4; ++q) {
        float4 acc = make_float4(0.f, 0.f, 0.f, 0.f);
        #pragma unroll
        for (int k = 0; k < K_FRAG; ++k) {
            const float4 v = *reinterpret_cast<const float4*>(row[k] + q * 4);
            const float wk = w[k];
            acc.x += wk * v.x;
            acc.y += wk * v.y;
            acc.z += wk * v.z;
            acc.w += wk * v.w;
        }
        const int c  = q * 4;                                  // unroll-constant
        float*    ob = (c < C_CH / 2) ? op_lo : op_hi;         // folds at compile time
        const int cc = (c < C_CH / 2) ? c : c - C_CH / 2;
        __builtin_nontemporal_store(acc.x * inv, ob + (size_t)(cc + 0) * HW_SZ);
        __builtin_nontemporal_store(acc.y * inv, ob + (size_t)(cc + 1) * HW_SZ);
        __builtin_nontemporal_store(acc.z * inv, ob + (size_t)(cc + 2) * HW_SZ);
        __builtin_nontemporal_store(acc.w * inv, ob + (size_t)(cc + 3) * HW_SZ);
    }
}

// --------------------------------------------------------------------------
// Fallback (workspace too small for the transposed table): gather directly
// from the (C,P) layout. Correct but strided; only used if ws_size is tiny.
// --------------------------------------------------------------------------
__global__ __launch_bounds__(256) void nwc_composite_fallback(
    const int*   __restrict__ frags,
    const float* __restrict__ alphas,
    const float* __restrict__ ptclds,  // (C, P) original layout
    float*       __restrict__ out,
    int NHW, int P)
{
    const int pix = blockIdx.x * blockDim.x + threadIdx.x;
    if (pix >= NHW) return;
    const int n  = pix >> 16;
    const int hw = pix & (HW_SZ - 1);

    const size_t base = ((size_t)n * K_FRAG) * HW_SZ + hw;

    int   idx[K_FRAG];
    float w[K_FRAG];
    float wsum = 0.0f;
    #pragma unroll
    for (int k = 0; k < K_FRAG; ++k) {
        const int   f = frags[base + (size_t)k * HW_SZ];
        const float a = alphas[base + (size_t)k * HW_SZ];
        const bool  m = (f >= 0);
        w[k]   = m ? a : 0.0f;
        idx[k] = m ? f : 0;
        wsum  += w[k];
    }
    const float inv = 1.0f / fmaxf(wsum, EPS);
    float* op = out + (size_t)n * C_CH * HW_SZ + hw;

    for (int c = 0; c < C_CH; ++c) {
        const float* rowp = ptclds + (size_t)c * P;
        float acc = 0.0f;
        #pragma unroll
        for (int k = 0; k < K_FRAG; ++k)
            acc += w[k] * rowp[idx[k]];
        __builtin_nontemporal_store(acc * inv, op + (size_t)c * HW_SZ);
    }
}

// --------------------------------------------------------------------------
// Harness entry point.
//   d_in[0] fragments (int, N*K*H*W), d_in[1] alphas (f32, same),
//   d_in[2] ptclds (f32, C*P). d_out: (N,C,H,W) f32.
// --------------------------------------------------------------------------
extern "C" void kernel_launch(void* const* d_in, const int* in_sizes, int n_in,
                              void* d_out, int out_size, void* d_ws, size_t ws_size,
                              hipStream_t stream)
{
    const int*   frags  = (const int*)  d_in[0];
    const float* alphas = (const float*)d_in[1];
    const float* ptclds = (const float*)d_in[2];
    float*       out    = (float*)      d_out;

    const int P   = in_sizes[2] / C_CH;           // 100000
    const int NHW = in_sizes[0] / K_FRAG;         // 4*256*256 = 262144

    const size_t tableBytes = (size_t)P * C_CH * sizeof(float);

    if (ws_size >= tableBytes) {
        float* tpt = (float*)d_ws;
        const int tBlocks = (P + 63) / 64;
        nwc_transpose_ptclds<<<tBlocks, 256, 0, stream>>>(ptclds, tpt, P);
        const int cBlocks = (NHW + 255) / 256;
        nwc_composite<<<cBlocks, 256, 0, stream>>>(frags, alphas, tpt, out, NHW);
    } else {
        const int cBlocks = (NHW + 255) / 256;
        nwc_composite_fallback<<<cBlocks, 256, 0, stream>>>(frags, alphas, ptclds,
                                                            out, NHW, P);
    }
}